// SparseMOEGPT_65833258713874
// MI455X (gfx1250) — compile-verified
//
#include <hip/hip_runtime.h>

#define ALPHA 1.702f
#define LIMIT 7.0f
#define NE 32
#define NK 4
#define NB 64
#define NH 2880
#define NI 2880
#define NF 5760 /* 2*I */

/* workspace layout */
#define WS_NM   0                 /* int: number of m-tiles            */
#define WS_MT   8                 /* int[64]: (expert<<16)|startRow    */
#define WS_CNT  72                /* int[32]: assignments per expert   */
#define WS_LIST 128               /* int[32*256]: assignment ids       */
#define WS_ACT_F  16384           /* float[256*NI]: activations        */
#define WS_EOUT_F (16384 + 256*NI)/* float[256*NH]: expert outputs     */

typedef __attribute__((ext_vector_type(2))) float v2f;
typedef __attribute__((ext_vector_type(8))) float v8f;

static __device__ __forceinline__ v8f wmma_k4(v2f a, v2f b, v8f c) {
  /* D = A(16x4 f32) * B(4x16 f32) + C(16x16 f32) */
  return __builtin_amdgcn_wmma_f32_16x16x4_f32(false, a, false, b,
                                               (short)0, c, false, false);
}

/* ---- 1. routing: bucket (b,k) assignments by expert, build m-tile list ---- */
__global__ void moe_route(const int* __restrict__ topk_idx, int* __restrict__ wsi) {
  if (threadIdx.x != 0 || blockIdx.x != 0) return;
  int cnt[NE];
  for (int e = 0; e < NE; ++e) cnt[e] = 0;
  for (int a = 0; a < NB * NK; ++a) {       /* fixed order => deterministic */
    int e = topk_idx[a];
    wsi[WS_LIST + e * 256 + cnt[e]] = a;    /* a = b*4+k */
    cnt[e]++;
  }
  int nm = 0;
  for (int e = 0; e < NE; ++e) {
    wsi[WS_CNT + e] = cnt[e];
    for (int s = 0; s < cnt[e]; s += 16) wsi[WS_MT + nm++] = (e << 16) | s;
  }
  wsi[WS_NM] = nm;
}

/* ---- 2. GEMM1 + clamped SiLU-GLU: act[aid, i] ---- */
__global__ void __launch_bounds__(256)
moe_gemm1(const float* __restrict__ x, const float* __restrict__ w_gu,
          const float* __restrict__ b_gu, float* __restrict__ ws) {
  const int* wsi = (const int*)ws;
  int mt = blockIdx.y;
  if (mt >= wsi[WS_NM]) return;
  int ent = wsi[WS_MT + mt];
  int e = ent >> 16, start = ent & 0xffff;
  int cnte = wsi[WS_CNT + e];

  __shared__ int s_aid[16];
  if (threadIdx.x < 16) {
    int r = start + (int)threadIdx.x;
    s_aid[threadIdx.x] = (r < cnte) ? wsi[WS_LIST + e * 256 + r] : -1;
  }
  __syncthreads();

  int wave = threadIdx.x >> 5, lane = threadIdx.x & 31;
  int itile = blockIdx.x * 8 + wave;
  if (itile * 16 >= NI) return;
  int i0 = itile * 16;
  int n = lane & 15;           /* column within tile / A row for loads */
  int khalf = lane >> 4;       /* K half per WMMA fp32 layout          */

  int aidA = s_aid[n];
  int tok = (aidA < 0) ? 0 : (aidA >> 2);
  const float* arow = x + (size_t)tok * NH;
  const float* wb = w_gu + (size_t)e * NH * NF + (size_t)(2 * (i0 + n));
  float bg = b_gu[(size_t)e * NF + 2 * (i0 + n)];
  float bu = b_gu[(size_t)e * NF + 2 * (i0 + n) + 1];
  v8f cg = {bg, bg, bg, bg, bg, bg, bg, bg};
  v8f cu = {bu, bu, bu, bu, bu, bu, bu, bu};

#pragma unroll 4
  for (int h = 0; h < NH; h += 4) {
    int hk = h + 2 * khalf;
    v2f a = *(const v2f*)(arow + hk);
    v2f p0 = *(const v2f*)(wb + (size_t)hk * NF);        /* gate,up @ row hk   */
    v2f p1 = *(const v2f*)(wb + (size_t)(hk + 1) * NF);  /* gate,up @ row hk+1 */
    int hp = hk + 64; if (hp >= NH) hp = hk;
    __builtin_prefetch(wb + (size_t)hp * NF, 0, 0);
    v2f bgv; bgv.x = p0.x; bgv.y = p1.x;
    v2f buv; buv.x = p0.y; buv.y = p1.y;
    cg = wmma_k4(a, bgv, cg);
    cu = wmma_k4(a, buv, cu);
  }

  float* act = ws + WS_ACT_F;
#pragma unroll
  for (int g = 0; g < 8; ++g) {
    int M = g + khalf * 8;
    int said = s_aid[M];
    float gate = fminf(cg[g], LIMIT);
    float up = fminf(fmaxf(cu[g], -LIMIT), LIMIT);
    float glu = gate / (1.0f + __expf(-ALPHA * gate));
    float v = (up + 1.0f) * glu;
    if (said >= 0) act[(size_t)said * NI + i0 + n] = v;
  }
}

/* ---- 3. GEMM2: eout[aid, h] = act[aid,:] . W_dn[e] + b_dn[e] ---- */
__global__ void __launch_bounds__(256)
moe_gemm2(const float* __restrict__ w_dn, const float* __restrict__ b_dn,
          float* __restrict__ ws) {
  const int* wsi = (const int*)ws;
  int mt = blockIdx.y;
  if (mt >= wsi[WS_NM]) return;
  int ent = wsi[WS_MT + mt];
  int e = ent >> 16, start = ent & 0xffff;
  int cnte = wsi[WS_CNT + e];

  __shared__ int s_aid[16];
  if (threadIdx.x < 16) {
    int r = start + (int)threadIdx.x;
    s_aid[threadIdx.x] = (r < cnte) ? wsi[WS_LIST + e * 256 + r] : -1;
  }
  __syncthreads();

  int wave = threadIdx.x >> 5, lane = threadIdx.x & 31;
  int htile = blockIdx.x * 8 + wave;
  if (htile * 16 >= NH) return;
  int h0 = htile * 16;
  int n = lane & 15, khalf = lane >> 4;

  const float* act = ws + WS_ACT_F;
  int aidA = s_aid[n];
  int rowa = (aidA < 0) ? 0 : aidA;
  const float* arow = act + (size_t)rowa * NI;
  const float* wb = w_dn + (size_t)e * NI * NH + (size_t)(h0 + n);
  float bd = b_dn[(size_t)e * NH + h0 + n];
  v8f c = {bd, bd, bd, bd, bd, bd, bd, bd};

#pragma unroll 4
  for (int i = 0; i < NI; i += 4) {
    int ik = i + 2 * khalf;
    v2f a = *(const v2f*)(arow + ik);
    v2f b;
    b.x = wb[(size_t)ik * NH];
    b.y = wb[(size_t)(ik + 1) * NH];
    int ip = ik + 64; if (ip >= NI) ip = ik;
    __builtin_prefetch(wb + (size_t)ip * NH, 0, 0);
    c = wmma_k4(a, b, c);
  }

  float* eout = ws + WS_EOUT_F;
#pragma unroll
  for (int g = 0; g < 8; ++g) {
    int M = g + khalf * 8;
    int said = s_aid[M];
    if (said >= 0) eout[(size_t)said * NH + h0 + n] = c[g];
  }
}

/* ---- 4. combine: out[b,h] = sum_k score[b,k] * eout[b*4+k, h] ---- */
__global__ void moe_combine(const float* __restrict__ scores,
                            const float* __restrict__ ws,
                            float* __restrict__ out) {
  int t = blockIdx.x * blockDim.x + threadIdx.x;
  if (t >= NB * NH) return;
  int b = t / NH, h = t - b * NH;
  const float* eout = ws + WS_EOUT_F;
  float s = 0.f;
#pragma unroll
  for (int k = 0; k < NK; ++k)
    s += scores[b * NK + k] * eout[(size_t)(b * NK + k) * NH + h];
  out[t] = s;
}

extern "C" void kernel_launch(void* const* d_in, const int* in_sizes, int n_in,
                              void* d_out, int out_size, void* d_ws, size_t ws_size,
                              hipStream_t stream) {
  const float* x      = (const float*)d_in[0];
  const int*   idx    = (const int*)d_in[1];
  const float* scores = (const float*)d_in[2];
  const float* w_gu   = (const float*)d_in[3];
  const float* b_gu   = (const float*)d_in[4];
  const float* w_dn   = (const float*)d_in[5];
  const float* b_dn   = (const float*)d_in[6];
  float* out = (float*)d_out;
  float* ws  = (float*)d_ws;
  int*   wsi = (int*)d_ws;

  moe_route<<<1, 32, 0, stream>>>(idx, wsi);

  dim3 g1((NI / 16 + 7) / 8, 64);   /* 23 x 64; y blocks past nMtiles exit */
  moe_gemm1<<<g1, 256, 0, stream>>>(x, w_gu, b_gu, ws);

  dim3 g2((NH / 16 + 7) / 8, 64);
  moe_gemm2<<<g2, 256, 0, stream>>>(w_dn, b_dn, ws);

  moe_combine<<<(NB * NH + 255) / 256, 256, 0, stream>>>(scores, ws, out);
}